// GCN_20890720928308
// MI455X (gfx1250) — compile-verified
//
#include <hip/hip_runtime.h>

#define N_NODES  100000
#define N_EDGES  1280000
#define D        64
#define N_GRAPHS 512
#define NTILES   (N_NODES / 16)   // 6250 exactly
#define WPAD     68               // LDS column pad -> conflict-free b64 B-fragment loads

typedef float v2f __attribute__((ext_vector_type(2)));
typedef float v8f __attribute__((ext_vector_type(8)));

// ---------------- degree / CSR build ----------------

__global__ void k_zero_counts(int* __restrict__ counts) {
  int i = blockIdx.x * blockDim.x + threadIdx.x;
  if (i < N_NODES) counts[i] = 0;
}

__global__ void k_count(const int* __restrict__ dst, int* __restrict__ counts) {
  int e = blockIdx.x * blockDim.x + threadIdx.x;
  if (e < N_EDGES) atomicAdd(&counts[dst[e]], 1);
}

__global__ __launch_bounds__(256)
void k_scan_block(const int* __restrict__ counts, int* __restrict__ rowptr,
                  int* __restrict__ bsum) {
  __shared__ int sh[256];
  int tid = threadIdx.x;
  int i = blockIdx.x * 256 + tid;
  int v = (i < N_NODES) ? counts[i] : 0;
  sh[tid] = v;
  __syncthreads();
  for (int off = 1; off < 256; off <<= 1) {
    int t = (tid >= off) ? sh[tid - off] : 0;
    __syncthreads();
    sh[tid] += t;
    __syncthreads();
  }
  if (i < N_NODES) rowptr[i] = sh[tid] - v;       // exclusive within block
  if (tid == 255) bsum[blockIdx.x] = sh[255];     // block total
}

__global__ __launch_bounds__(512)
void k_scan_bsum(int* __restrict__ bsum, int nb) {
  __shared__ int sh[512];
  int tid = threadIdx.x;
  int v = (tid < nb) ? bsum[tid] : 0;
  sh[tid] = v;
  __syncthreads();
  for (int off = 1; off < 512; off <<= 1) {
    int t = (tid >= off) ? sh[tid - off] : 0;
    __syncthreads();
    sh[tid] += t;
    __syncthreads();
  }
  if (tid < nb) bsum[tid] = sh[tid] - v;          // exclusive across blocks
}

__global__ __launch_bounds__(256)
void k_finalize(const int* __restrict__ counts, int* __restrict__ rowptr,
                const int* __restrict__ bsum, int* __restrict__ cursor,
                float* __restrict__ dinv) {
  int i = blockIdx.x * 256 + threadIdx.x;
  if (i < N_NODES) {
    int r = rowptr[i] + bsum[blockIdx.x];
    rowptr[i] = r;
    cursor[i] = r;
    dinv[i] = rsqrtf((float)counts[i] + 1.0f);    // +1 self-loop
  }
  if (i == 0) rowptr[N_NODES] = N_EDGES;
}

__global__ void k_scatter(const int* __restrict__ src, const int* __restrict__ dst,
                          int* __restrict__ cursor, int* __restrict__ csr_src) {
  int e = blockIdx.x * blockDim.x + threadIdx.x;
  if (e < N_EDGES) {
    int pos = atomicAdd(&cursor[dst[e]], 1);
    csr_src[pos] = src[e];
  }
}

// ---------------- fp32 WMMA matmul: H[N_NODES,64] = X[N_NODES,64] @ W[64,64] ----------------
// One wave computes a 16-row x 64-col tile with V_WMMA_F32_16X16X4_F32.
// A-frag (16x4, MxK): lane holds row M=lane&15; VGPR0/1 hold K = 2*(lane>>4), +1.
// B-frag (4x16, KxN): lane holds col N=lane&15; VGPR0/1 hold K = 2*(lane>>4), +1.
// W stored transposed in LDS (sw[n*WPAD + k]) so each B fragment is one ds_load_b64;
// pad 68 makes lane word-addresses 4*(lane&15)+2*(lane>>4) mod 64 -> all 64 banks, no conflict.
__global__ __launch_bounds__(256)
void k_matmul(const float* __restrict__ X, const float* __restrict__ W,
              float* __restrict__ H) {
  __shared__ __align__(16) float sw[D * WPAD];
  int tid = threadIdx.x;
  for (int i = tid; i < D * D; i += 256) {
    int k = i >> 6;
    int n = i & 63;
    sw[n * WPAD + k] = W[i];
  }
  __syncthreads();

  int wave = tid >> 5;
  int lane = tid & 31;
  int tile = blockIdx.x * 8 + wave;
  if (tile >= NTILES) return;           // wave-uniform: EXEC stays all-1 in WMMA path

  int m  = lane & 15;
  int kh = lane >> 4;                   // 0 or 1
  long row0 = (long)tile * 16;
  const float* xrow = X + (row0 + m) * D;

  v8f acc0 = {}, acc1 = {}, acc2 = {}, acc3 = {};
#pragma unroll
  for (int kk = 0; kk < D; kk += 4) {
    int ka = kk + 2 * kh;
    v2f a   = *(const v2f*)&xrow[ka];
    v2f b0v = *(const v2f*)&sw[(m +  0) * WPAD + ka];
    v2f b1v = *(const v2f*)&sw[(m + 16) * WPAD + ka];
    v2f b2v = *(const v2f*)&sw[(m + 32) * WPAD + ka];
    v2f b3v = *(const v2f*)&sw[(m + 48) * WPAD + ka];
    acc0 = __builtin_amdgcn_wmma_f32_16x16x4_f32(false, a, false, b0v, (short)0, acc0, false, false);
    acc1 = __builtin_amdgcn_wmma_f32_16x16x4_f32(false, a, false, b1v, (short)0, acc1, false, false);
    acc2 = __builtin_amdgcn_wmma_f32_16x16x4_f32(false, a, false, b2v, (short)0, acc2, false, false);
    acc3 = __builtin_amdgcn_wmma_f32_16x16x4_f32(false, a, false, b3v, (short)0, acc3, false, false);
  }

  float* hbase = H + (row0 + 8 * kh) * D + m;
#pragma unroll
  for (int v = 0; v < 8; ++v) {
    hbase[(long)v * D +  0] = acc0[v];
    hbase[(long)v * D + 16] = acc1[v];
    hbase[(long)v * D + 32] = acc2[v];
    hbase[(long)v * D + 48] = acc3[v];
  }
}

// ---------------- gather aggregation + bias + ReLU ----------------
// One wave per node; lane owns features 2*lane, 2*lane+1.
// Lanes cooperatively load <=32 edges' (src, dinv[src]); each edge is broadcast
// with v_readlane (uniform j), and the wave fetches the 256B feature row with a
// single b64 load per lane. out = relu(dinv[i]*sum + h[i]*dinv^2 + b).
__global__ __launch_bounds__(256)
void k_aggregate(const float* __restrict__ H, const int* __restrict__ rowptr,
                 const int* __restrict__ csr_src, const float* __restrict__ dinv,
                 const float* __restrict__ bias, float* __restrict__ OUT) {
  int tid  = threadIdx.x;
  int lane = tid & 31;
  int node = blockIdx.x * 8 + (tid >> 5);
  if (node >= N_NODES) return;

  int beg = rowptr[node];
  int end = rowptr[node + 1];
  int f2  = lane * 2;

  float acc0 = 0.0f, acc1 = 0.0f;
  for (int base = beg; base < end; base += 32) {
    int cnt = end - base;
    if (cnt > 32) cnt = 32;
    int s = 0;
    int cbits = 0;
    if (lane < cnt) {
      s = csr_src[base + lane];
      cbits = (int)__float_as_uint(dinv[s]);
    }
    for (int j = 0; j < cnt; ++j) {              // j is wave-uniform
      int   sj = __builtin_amdgcn_readlane(s, j);
      float cj = __uint_as_float((unsigned)__builtin_amdgcn_readlane(cbits, j));
      v2f h = *(const v2f*)&H[(long)sj * D + f2];
      acc0 += h.x * cj;
      acc1 += h.y * cj;
    }
  }

  float di = dinv[node];
  v2f hs = *(const v2f*)&H[(long)node * D + f2];
  v2f bb = *(const v2f*)&bias[f2];
  float r0 = di * acc0 + hs.x * (di * di) + bb.x;
  float r1 = di * acc1 + hs.y * (di * di) + bb.y;
  v2f o;
  o.x = fmaxf(r0, 0.0f);
  o.y = fmaxf(r1, 0.0f);
  *(v2f*)&OUT[(long)node * D + f2] = o;
}

// ---------------- pooling + head ----------------

__global__ void k_zero_pool(float* __restrict__ gsum, float* __restrict__ gcnt) {
  int i = blockIdx.x * blockDim.x + threadIdx.x;
  if (i < N_GRAPHS * D) gsum[i] = 0.0f;
  if (i < N_GRAPHS)     gcnt[i] = 0.0f;
}

__global__ __launch_bounds__(256)
void k_pool(const float* __restrict__ F, const int* __restrict__ batch,
            float* __restrict__ gsum, float* __restrict__ gcnt) {
  int tid  = threadIdx.x;
  int node = blockIdx.x * 4 + (tid >> 6);
  int f    = tid & 63;
  if (node >= N_NODES) return;
  int g = batch[node];
  atomicAdd(&gsum[g * D + f], F[(long)node * D + f]);
  if (f == 0) atomicAdd(&gcnt[g], 1.0f);
}

__global__ void k_head(const float* __restrict__ gsum, const float* __restrict__ gcnt,
                       const float* __restrict__ fc_w, const float* __restrict__ fc_b,
                       float* __restrict__ out) {
  int g = blockIdx.x * blockDim.x + threadIdx.x;
  if (g >= N_GRAPHS) return;
  float c = fmaxf(gcnt[g], 1.0f);
  float s = 0.0f;
  for (int f = 0; f < D; ++f) s += gsum[g * D + f] * fc_w[f];
  out[g] = s / c + fc_b[0];
}

// ---------------- launch ----------------

extern "C" void kernel_launch(void* const* d_in, const int* in_sizes, int n_in,
                              void* d_out, int out_size, void* d_ws, size_t ws_size,
                              hipStream_t stream) {
  (void)in_sizes; (void)n_in; (void)out_size; (void)ws_size;

  const float* x     = (const float*)d_in[0];
  const int*   edge  = (const int*)d_in[1];
  const int*   src   = edge;
  const int*   dst   = edge + N_EDGES;
  const int*   batch = (const int*)d_in[2];
  const float* w0    = (const float*)d_in[3];
  const float* b0    = (const float*)d_in[4];
  const float* w1    = (const float*)d_in[5];
  const float* b1    = (const float*)d_in[6];
  const float* w2    = (const float*)d_in[7];
  const float* b2    = (const float*)d_in[8];
  const float* fc_w  = (const float*)d_in[9];
  const float* fc_b  = (const float*)d_in[10];
  float*       out   = (float*)d_out;

  // workspace carve-out (~58 MB)
  char* ws = (char*)d_ws;
  size_t off = 0;
  auto carve = [&](size_t bytes) -> void* {
    void* p = ws + off;
    off = (off + bytes + 255) & ~(size_t)255;
    return p;
  };
  float* bufA    = (float*)carve((size_t)N_NODES * D * sizeof(float));
  float* bufB    = (float*)carve((size_t)N_NODES * D * sizeof(float));
  float* dinv    = (float*)carve((size_t)N_NODES * sizeof(float));
  int*   counts  = (int*)  carve((size_t)N_NODES * sizeof(int));
  int*   rowptr  = (int*)  carve((size_t)(N_NODES + 1) * sizeof(int));
  int*   cursor  = (int*)  carve((size_t)N_NODES * sizeof(int));
  int*   csr_src = (int*)  carve((size_t)N_EDGES * sizeof(int));
  int*   bsum    = (int*)  carve(512 * sizeof(int));
  float* gsum    = (float*)carve((size_t)N_GRAPHS * D * sizeof(float));
  float* gcnt    = (float*)carve((size_t)N_GRAPHS * sizeof(float));

  const int nNodeBlk = (N_NODES + 255) / 256;   // 391
  const int nEdgeBlk = (N_EDGES + 255) / 256;   // 5000
  const int nMmBlk   = (NTILES + 7) / 8;        // 782
  const int nAggBlk  = N_NODES / 8;             // 12500 (wave per node)
  const int nPoolBlk = N_NODES / 4;             // 25000

  // CSR build (layer-invariant)
  k_zero_counts<<<nNodeBlk, 256, 0, stream>>>(counts);
  k_count      <<<nEdgeBlk, 256, 0, stream>>>(dst, counts);
  k_scan_block <<<nNodeBlk, 256, 0, stream>>>(counts, rowptr, bsum);
  k_scan_bsum  <<<1, 512, 0, stream>>>(bsum, nNodeBlk);
  k_finalize   <<<nNodeBlk, 256, 0, stream>>>(counts, rowptr, bsum, cursor, dinv);
  k_scatter    <<<nEdgeBlk, 256, 0, stream>>>(src, dst, cursor, csr_src);

  // layer 1
  k_matmul   <<<nMmBlk,  256, 0, stream>>>(x, w0, bufB);
  k_aggregate<<<nAggBlk, 256, 0, stream>>>(bufB, rowptr, csr_src, dinv, b0, bufA);
  // layer 2
  k_matmul   <<<nMmBlk,  256, 0, stream>>>(bufA, w1, bufB);
  k_aggregate<<<nAggBlk, 256, 0, stream>>>(bufB, rowptr, csr_src, dinv, b1, bufA);
  // layer 3
  k_matmul   <<<nMmBlk,  256, 0, stream>>>(bufA, w2, bufB);
  k_aggregate<<<nAggBlk, 256, 0, stream>>>(bufB, rowptr, csr_src, dinv, b2, bufA);

  // pool + head
  k_zero_pool<<<(N_GRAPHS * D + 255) / 256, 256, 0, stream>>>(gsum, gcnt);
  k_pool     <<<nPoolBlk, 256, 0, stream>>>(bufA, batch, gsum, gcnt);
  k_head     <<<(N_GRAPHS + 255) / 256, 256, 0, stream>>>(gsum, gcnt, fc_w, fc_b, out);
}